// GCN_46093589021048
// MI455X (gfx1250) — compile-verified
//
#include <hip/hip_runtime.h>

#define NN 10000      // nodes
#define NE 160000     // edges
#define FIN 512
#define NHID 512
#define LDS_STRIDE 516   // 512 + 1 DWORD pad per 128 DWORDs -> conflict-free banks

typedef float v2f __attribute__((ext_vector_type(2)));
typedef float v8f __attribute__((ext_vector_type(8)));
typedef unsigned int u32x4 __attribute__((ext_vector_type(4)));
typedef int i32x4 __attribute__((ext_vector_type(4)));
typedef int i32x8 __attribute__((ext_vector_type(8)));

#if defined(__has_builtin)
#if __has_builtin(__builtin_amdgcn_wmma_f32_16x16x4_f32)
#define HAVE_WMMA_F32 1
#endif
#if __has_builtin(__builtin_amdgcn_tensor_load_to_lds)
#define HAVE_TDM 1
#endif
#if __has_builtin(__builtin_amdgcn_s_wait_tensorcnt)
#define HAVE_WAIT_TENSOR 1
#endif
#endif

#if defined(__has_include)
#if __has_include(<hip/amd_detail/amd_gfx1250_TDM.h>)
#define TDM_6ARG 1     // therock-10.0 headers -> 6-arg builtin arity
#endif
#endif

__device__ __forceinline__ void wait_tensorcnt0() {
#if HAVE_WAIT_TENSOR
    __builtin_amdgcn_s_wait_tensorcnt(0);
#else
    asm volatile("s_wait_tensorcnt 0x0" ::: "memory");
#endif
}

// ---------------- degree / norm kernels ----------------
__global__ void k_init_deg(float* __restrict__ degw, float* __restrict__ degc) {
    int i = blockIdx.x * blockDim.x + threadIdx.x;
    if (i < NN) { degw[i] = 1.0f; degc[i] = 1.0f; }   // self-loop contributions
}

__global__ void k_edge_deg(const int* __restrict__ col, const float* __restrict__ ew,
                           float* __restrict__ degw, float* __restrict__ degc) {
    int e = blockIdx.x * blockDim.x + threadIdx.x;
    if (e < NE) {
        int c = col[e];
        atomicAdd(&degw[c], ew[e]);
        atomicAdd(&degc[c], 1.0f);
    }
}

__global__ void k_norms(const float* __restrict__ degw, const float* __restrict__ degc,
                        float* __restrict__ dis, float* __restrict__ dinv) {
    int i = blockIdx.x * blockDim.x + threadIdx.x;
    if (i < NN) {
        float d = degw[i];
        dis[i]  = (d > 0.0f) ? rsqrtf(d) : 0.0f;
        dinv[i] = 1.0f / fmaxf(degc[i], 1.0f);
    }
}

// ---------------- WMMA f32 GEMM: H0 = X (NN x FIN) @ W1^T ----------------
// Per block: TDM-stage the 16x512 X tile into LDS (bank-conflict-free padded),
// 8 waves each compute a 16x64 C tile; A from LDS (ds_load_b64), B from global.
#if HAVE_WMMA_F32
__global__ __launch_bounds__(256) void k_gemm_wmma(const float* __restrict__ X,
                                                   const float* __restrict__ W1,
                                                   float* __restrict__ H0) {
    __shared__ float tile[16 * LDS_STRIDE];

    const int lane = threadIdx.x & 31;
    const int wave = threadIdx.x >> 5;          // 8 waves
    const int m0 = blockIdx.x * 16;             // 625 blocks cover NN=10000
    const int nb = wave * 64;                   // each wave: 16x64 tile
    const int l15 = lane & 15;
    const int koff = (lane >> 4) << 1;          // lanes 0-15 -> K 0,1 ; lanes 16-31 -> K 2,3

#if HAVE_TDM
    if (threadIdx.x < 32) {                     // wave 0 issues the tensor DMA once
        unsigned long long ga = (unsigned long long)(const void*)(X + (size_t)m0 * FIN);
        u32x4 tg0;
        tg0.x = 1u;                                           // count=1, user mode
        tg0.y = (unsigned)(size_t)(void*)tile;                // lds_addr
        tg0.z = (unsigned)(ga & 0xFFFFFFFFull);               // global_addr[31:0]
        tg0.w = (unsigned)((ga >> 32) & 0x1FFFFFFull) |       // global_addr[56:32]
                0x80000000u;                                  // type=2 ("image")
        // data_size=4B(code 2), pad_enable, pad_interval=128 DW (code 6), pad_amount=1 DW (code 0)
        i32x8 tg1 = { (int)((2u << 16) | (1u << 20) | (6u << 22)),
                      (int)(512u << 16),     // tensor_dim0 = 512 (bits 79:48)
                      (int)(16u << 16),      // tensor_dim1 = 16  (bits 111:80)
                      (int)(512u << 16),     // tile_dim0   = 512 (bits 127:112)
                      16,                    // tile_dim1   = 16
                      512,                   // tensor_dim0_stride = 512
                      0, 0 };
        i32x4 tzero4 = { 0, 0, 0, 0 };
#if TDM_6ARG
        i32x8 tzero8 = { 0, 0, 0, 0, 0, 0, 0, 0 };
        __builtin_amdgcn_tensor_load_to_lds(tg0, tg1, tzero4, tzero4, tzero8, 0);
#else
        __builtin_amdgcn_tensor_load_to_lds(tg0, tg1, tzero4, tzero4, 0);
#endif
        wait_tensorcnt0();
    }
    __syncthreads();
#else
    // fallback: coalesced manual staging with the same padded layout
    for (int idx = threadIdx.x; idx < 16 * 128; idx += 256) {
        int r = idx >> 7;
        int c = (idx & 127) << 2;
        float4 v = *(const float4*)(X + (size_t)(m0 + r) * FIN + c);
        float* d = tile + r * LDS_STRIDE + c + (c >> 7);
        d[0] = v.x; d[1] = v.y; d[2] = v.z; d[3] = v.w;
    }
    __syncthreads();
#endif

    const float* __restrict__ ap = tile + l15 * LDS_STRIDE + koff;
    const float* __restrict__ w0 = W1 + (size_t)(nb +  0 + l15) * FIN + koff;
    const float* __restrict__ w1 = W1 + (size_t)(nb + 16 + l15) * FIN + koff;
    const float* __restrict__ w2 = W1 + (size_t)(nb + 32 + l15) * FIN + koff;
    const float* __restrict__ w3 = W1 + (size_t)(nb + 48 + l15) * FIN + koff;

    v8f acc0 = {}, acc1 = {}, acc2 = {}, acc3 = {};
#pragma unroll 2
    for (int k = 0; k < FIN; k += 4) {
        v2f a  = *(const v2f*)(ap + k + (k >> 7));   // LDS, pad-aware, bank-conflict-free
        v2f b0 = *(const v2f*)(w0 + k);
        v2f b1 = *(const v2f*)(w1 + k);
        v2f b2 = *(const v2f*)(w2 + k);
        v2f b3 = *(const v2f*)(w3 + k);
        acc0 = __builtin_amdgcn_wmma_f32_16x16x4_f32(false, a, false, b0, (short)0, acc0, false, false);
        acc1 = __builtin_amdgcn_wmma_f32_16x16x4_f32(false, a, false, b1, (short)0, acc1, false, false);
        acc2 = __builtin_amdgcn_wmma_f32_16x16x4_f32(false, a, false, b2, (short)0, acc2, false, false);
        acc3 = __builtin_amdgcn_wmma_f32_16x16x4_f32(false, a, false, b3, (short)0, acc3, false, false);
    }

    // C/D layout: VGPR r -> row m0 + r + 8*(lane>=16), col = lane&15
    const int rbase = m0 + ((lane >> 4) << 3);
#pragma unroll
    for (int r = 0; r < 8; ++r) {
        float* o = H0 + (size_t)(rbase + r) * NHID + nb + l15;
        o[0]  = acc0[r];
        o[16] = acc1[r];
        o[32] = acc2[r];
        o[48] = acc3[r];
    }
}
#else
__global__ __launch_bounds__(256) void k_gemm_naive(const float* __restrict__ X,
                                                    const float* __restrict__ W1,
                                                    float* __restrict__ H0) {
    size_t idx = (size_t)blockIdx.x * blockDim.x + threadIdx.x;
    if (idx >= (size_t)NN * NHID) return;
    int i = (int)(idx >> 9), j = (int)(idx & (NHID - 1));
    const float* xr = X + (size_t)i * FIN;
    const float* wr = W1 + (size_t)j * FIN;
    float s = 0.f;
    for (int k = 0; k < FIN; ++k) s = fmaf(xr[k], wr[k], s);
    H0[idx] = s;
}
#endif

// ---------------- self-loop term: hacc = dis[i]^2 * h0 ----------------
__global__ void k_selfloop(const float* __restrict__ h0, const float* __restrict__ dis,
                           float* __restrict__ hacc) {
    size_t idx = (size_t)blockIdx.x * blockDim.x + threadIdx.x;
    if (idx < (size_t)NN * NHID) {
        int i = (int)(idx >> 9);
        float s = dis[i];
        hacc[idx] = s * s * h0[idx];
    }
}

// ---------------- layer-1 edge aggregation: one block per edge ----------------
__global__ __launch_bounds__(256) void k_edge_agg(const int* __restrict__ row,
                                                  const int* __restrict__ col,
                                                  const float* __restrict__ ew,
                                                  const float* __restrict__ dis,
                                                  const float* __restrict__ h0,
                                                  float* __restrict__ hacc) {
    int e = blockIdx.x;
    int r = row[e], c = col[e];
    float nrm = dis[r] * ew[e] * dis[c];
    int f = threadIdx.x << 1;                  // 256 threads x float2 = 512 feats
    float2 hv = *(const float2*)(h0 + (size_t)r * NHID + f);
    float* dst = hacc + (size_t)c * NHID + f;
    atomicAdd(dst + 0, nrm * hv.x);
    atomicAdd(dst + 1, nrm * hv.y);
}

// ---------------- ReLU+bias -> features; per-node dots with w2_out / w2_root ----------------
__global__ __launch_bounds__(256) void k_relu_dots(const float* __restrict__ hacc,
                                                   const float* __restrict__ b1,
                                                   const float* __restrict__ w2o,
                                                   const float* __restrict__ w2r,
                                                   float* __restrict__ feat,
                                                   float* __restrict__ p,
                                                   float* __restrict__ q,
                                                   float* __restrict__ outacc) {
    int i = blockIdx.x;
    int f = threadIdx.x << 1;
    float2 v = *(const float2*)(hacc + (size_t)i * NHID + f);
    float2 b = *(const float2*)(b1 + f);
    float h0v = fmaxf(v.x + b.x, 0.0f);
    float h1v = fmaxf(v.y + b.y, 0.0f);
    *(float2*)(feat + (size_t)i * NHID + f) = make_float2(h0v, h1v);

    float2 wo = *(const float2*)(w2o + f);
    float2 wr = *(const float2*)(w2r + f);
    float pl = h0v * wo.x + h1v * wo.y;
    float ql = h0v * wr.x + h1v * wr.y;

    __shared__ float sp[256];
    __shared__ float sq[256];
    sp[threadIdx.x] = pl;
    sq[threadIdx.x] = ql;
    __syncthreads();
#pragma unroll
    for (int s = 128; s > 0; s >>= 1) {
        if ((int)threadIdx.x < s) {
            sp[threadIdx.x] += sp[threadIdx.x + s];
            sq[threadIdx.x] += sq[threadIdx.x + s];
        }
        __syncthreads();
    }
    if (threadIdx.x == 0) {
        float pv = sp[0];
        p[i] = pv;
        q[i] = sq[0];
        outacc[i] = pv;                        // self-loop term of layer-2 agg
    }
}

// ---------------- layer-2 scalar edge scatter ----------------
__global__ void k_edge_scalar(const int* __restrict__ row, const int* __restrict__ col,
                              const float* __restrict__ p, float* __restrict__ outacc) {
    int e = blockIdx.x * blockDim.x + threadIdx.x;
    if (e < NE) atomicAdd(&outacc[col[e]], p[row[e]]);
}

__global__ void k_finalize(const float* __restrict__ outacc, const float* __restrict__ dinv,
                           const float* __restrict__ q, const float* __restrict__ b2,
                           float* __restrict__ out) {
    int i = blockIdx.x * blockDim.x + threadIdx.x;
    if (i < NN) out[i] = dinv[i] * outacc[i] + b2[0] + q[i];
}

extern "C" void kernel_launch(void* const* d_in, const int* in_sizes, int n_in,
                              void* d_out, int out_size, void* d_ws, size_t ws_size,
                              hipStream_t stream) {
    (void)in_sizes; (void)n_in; (void)out_size; (void)ws_size;

    const float* x   = (const float*)d_in[0];
    const int*   ei  = (const int*)  d_in[1];   // [2, E] : row then col
    const float* ew  = (const float*)d_in[2];
    const float* W1  = (const float*)d_in[3];
    const float* b1  = (const float*)d_in[4];
    const float* W2o = (const float*)d_in[5];
    const float* b2  = (const float*)d_in[6];
    const float* W2r = (const float*)d_in[7];
    const int* row = ei;
    const int* col = ei + NE;

    float* out  = (float*)d_out;               // [NN]
    float* feat = (float*)d_out + NN;          // [NN * NHID]

    float* ws     = (float*)d_ws;
    float* h0     = ws;                              // NN*NHID
    float* hacc   = h0   + (size_t)NN * NHID;        // NN*NHID
    float* degw   = hacc + (size_t)NN * NHID;        // NN
    float* degc   = degw + NN;                       // NN
    float* dis    = degc + NN;                       // NN
    float* dinv   = dis  + NN;                       // NN
    float* p      = dinv + NN;                       // NN
    float* q      = p    + NN;                       // NN
    float* outacc = q    + NN;                       // NN

    k_init_deg<<<(NN + 255) / 256, 256, 0, stream>>>(degw, degc);
    k_edge_deg<<<(NE + 255) / 256, 256, 0, stream>>>(col, ew, degw, degc);
    k_norms<<<(NN + 255) / 256, 256, 0, stream>>>(degw, degc, dis, dinv);
#if HAVE_WMMA_F32
    k_gemm_wmma<<<NN / 16, 256, 0, stream>>>(x, W1, h0);
#else
    k_gemm_naive<<<(unsigned)(((size_t)NN * NHID + 255) / 256), 256, 0, stream>>>(x, W1, h0);
#endif
    k_selfloop<<<(unsigned)(((size_t)NN * NHID + 255) / 256), 256, 0, stream>>>(h0, dis, hacc);
    k_edge_agg<<<NE, 256, 0, stream>>>(row, col, ew, dis, h0, hacc);
    k_relu_dots<<<NN, 256, 0, stream>>>(hacc, b1, W2o, W2r, feat, p, q, outacc);
    k_edge_scalar<<<(NE + 255) / 256, 256, 0, stream>>>(row, col, p, outacc);
    k_finalize<<<(NN + 255) / 256, 256, 0, stream>>>(outacc, dinv, q, b2, out);
}